// TopoLoss_43353399886340
// MI455X (gfx1250) — compile-verified
//
#include <hip/hip_runtime.h>
#include <hip/hip_bf16.h>
#include <stdint.h>

#ifndef __has_builtin
#define __has_builtin(x) 0
#endif

#define K_PD    128
#define NPIX    65536            // 256*256
#define THREADS 1024
#define PPT     (NPIX / THREADS) // 64 pixels per thread
#define BIGF    1e9f
#define HBIGF   5e8f             // BIG/2 validity threshold, matches reference

// ---------- gfx1250 async global<->LDS paths (guarded; fallbacks = plain ops) ---
#if __has_builtin(__builtin_amdgcn_global_load_async_to_lds_b128)
#define ASYNC_LDS 1
#else
#define ASYNC_LDS 0
#endif
#if __has_builtin(__builtin_amdgcn_global_store_async_from_lds_b128)
#define ASYNC_STORE 1
#else
#define ASYNC_STORE 0
#endif
#if __has_builtin(__builtin_amdgcn_s_wait_asynccnt)
#define WAIT_ASYNC() __builtin_amdgcn_s_wait_asynccnt(0)
#else
#define WAIT_ASYNC() asm volatile("s_wait_asynccnt 0" ::: "memory")
#endif

typedef int v4i __attribute__((ext_vector_type(4)));

__device__ __forceinline__ __attribute__((address_space(1))) v4i*
to_global_v4(const void* p) {
    // ptr -> int -> AS1 ptr (global generic VA == AS1 VA)
    return (__attribute__((address_space(1))) v4i*)(unsigned long long)p;
}
__device__ __forceinline__ __attribute__((address_space(3))) v4i*
to_lds_v4(void* p) {
    // low 32 bits of a generic LDS pointer are the LDS byte offset
    return (__attribute__((address_space(3))) v4i*)(unsigned int)(unsigned long long)p;
}

// ---------- order-preserving float<->uint key mapping ---------------------------
__device__ __forceinline__ unsigned fkey(float f, int dir) {
    unsigned u = __float_as_uint(f);
    u ^= (u & 0x80000000u) ? 0xFFFFFFFFu : 0x80000000u; // monotone map
    return dir ? ~u : u;                                 // dir=1 -> descending
}
__device__ __forceinline__ float unfkey(unsigned k, int dir) {
    if (dir) k = ~k;
    k = (k & 0x80000000u) ? (k ^ 0x80000000u) : ~k;
    return __uint_as_float(k);
}

// ---------- per-block top-K (sorted) selection via 8-bit radix select ----------
// simg: LDS image (65536 f32). mask: LDS bitmask (2048 words). dir: 0 asc, 1 desc.
// out: 128 floats (global), padded with `sentinel` past the subset size.
// hist8: 8 wave-banked 256-bin histograms; hist/pfx: combined bins + prefix sums.
__device__ void topk_select(const float* simg, const unsigned* mask, int dir,
                            float sentinel, float* out,
                            unsigned* hist8, unsigned* hist, unsigned* pfx,
                            unsigned* cand, unsigned* ctrl, int tid)
{
    // ctrl[0]=need/eqTake  ctrl[1]=total  ctrl[2]=takeAll  ctrl[3]=prefix
    // ctrl[4]=less-counter ctrl[5]=eq-counter
    if (tid == 0) { ctrl[0] = K_PD; ctrl[1] = 0; ctrl[2] = 0; ctrl[3] = 0; }
    __syncthreads();

    const int base = tid * PPT;
    const unsigned m0 = mask[2 * tid], m1 = mask[2 * tid + 1];
    const int hb = ((tid >> 5) & 7) << 8;        // histogram bank of wave % 8 (4 waves share one)

    for (int pass = 0; pass < 4; ++pass) {
        const unsigned takeAll = ctrl[2];
        const unsigned prefix  = ctrl[3];
        const unsigned need    = ctrl[0];
        hist8[tid] = 0; hist8[tid + 1024] = 0;   // clear all 8 banks
        __syncthreads();

        const int sh = 24 - 8 * pass;
        if (!takeAll) {
            for (int k = 0; k < PPT; ++k) {
                const unsigned bit = (k < 32) ? ((m0 >> k) & 1u) : ((m1 >> (k - 32)) & 1u);
                if (bit) {
                    const unsigned key = fkey(simg[base + k], dir);
                    if (pass == 0 || (key >> (sh + 8)) == prefix)
                        atomicAdd(&hist8[hb + ((key >> sh) & 255u)], 1u);
                }
            }
        }
        __syncthreads();

        // combine banks
        if (tid < 256) {
            unsigned h = 0;
#pragma unroll
            for (int c = 0; c < 8; ++c) h += hist8[(c << 8) + tid];
            hist[tid] = h;
            pfx[tid]  = h;
        }
        __syncthreads();

        // inclusive prefix sum over 256 bins (Hillis-Steele, 8 steps)
        for (int off = 1; off < 256; off <<= 1) {
            unsigned v = 0;
            if (tid < 256 && tid >= off) v = pfx[tid - off];
            __syncthreads();
            if (tid < 256 && tid >= off) pfx[tid] += v;
            __syncthreads();
        }

        // parallel bin pick: the unique bin with exc < need <= inc
        if (!takeAll && tid < 256) {
            const unsigned inc = pfx[tid];
            const unsigned exc = inc - hist[tid];
            if (pass == 0 && tid == 255) {
                ctrl[1] = inc;                   // subset size
                if (inc < K_PD) ctrl[2] = 1;     // fewer than K -> take everything
            }
            if (exc < need && need <= inc) {
                ctrl[0] = need - exc;            // rank within chosen bin (>=1)
                ctrl[3] = (prefix << 8) | (unsigned)tid;
            }
        }
        __syncthreads();
    }

    const unsigned takeAll = ctrl[2];
    const unsigned total   = ctrl[1];
    const unsigned T       = ctrl[3];            // key of the K-th smallest
    const unsigned eqTake  = ctrl[0];
    const unsigned cntLess = K_PD - eqTake;      // #keys strictly < T (when !takeAll)

    if (tid == 0) { ctrl[4] = 0; ctrl[5] = 0; }
    __syncthreads();

    for (int k = 0; k < PPT; ++k) {
        const unsigned bit = (k < 32) ? ((m0 >> k) & 1u) : ((m1 >> (k - 32)) & 1u);
        if (bit) {
            const unsigned key = fkey(simg[base + k], dir);
            if (takeAll) {
                const unsigned pos = atomicAdd(&ctrl[4], 1u);
                cand[pos] = key;
            } else if (key < T) {
                const unsigned pos = atomicAdd(&ctrl[4], 1u);
                cand[pos] = key;                 // pos < cntLess <= 127 by construction
            } else if (key == T) {
                const unsigned e = atomicAdd(&ctrl[5], 1u);
                if (e < eqTake) cand[cntLess + e] = key;
            }
        }
    }
    __syncthreads();

    const unsigned valid = takeAll ? total : (unsigned)K_PD;
    if ((unsigned)tid >= valid && tid < K_PD) cand[tid] = 0xFFFFFFFFu;
    __syncthreads();

    // bitonic sort of 128 keys, ascending; 64 worker lanes
    for (unsigned k2 = 2; k2 <= (unsigned)K_PD; k2 <<= 1) {
        for (unsigned j = k2 >> 1; j > 0; j >>= 1) {
            if (tid < K_PD / 2) {
                const unsigned w  = (unsigned)tid;
                const unsigned e1 = 2u * j * (w / j) + (w & (j - 1u));
                const unsigned e2 = e1 + j;
                const bool up = ((e1 & k2) == 0u);
                const unsigned a = cand[e1], b = cand[e2];
                if ((a > b) == up) { cand[e1] = b; cand[e2] = a; }
            }
            __syncthreads();
        }
    }

#if ASYNC_STORE
    // unmap in place, then DMA the 512B list LDS->global (async store path)
    if (tid < K_PD) {
        const float v = ((unsigned)tid < valid) ? unfkey(cand[tid], dir) : sentinel;
        cand[tid] = __float_as_uint(v);
    }
    __syncthreads();
    if (tid < K_PD / 4) {                        // 32 lanes x 16B = 512B
        __builtin_amdgcn_global_store_async_from_lds_b128(
            to_global_v4((const char*)out + tid * 16),
            to_lds_v4((char*)cand + tid * 16), 0, 0);
    }
    WAIT_ASYNC();                                // cand is reused by the next select
    __syncthreads();
#else
    if (tid < K_PD) {
        out[tid] = ((unsigned)tid < valid) ? unfkey(cand[tid], dir) : sentinel;
    }
    __syncthreads();
#endif
}

// ---------- kernel A: one image per block; 4 sorted top-K lists -> ws ----------
__global__ __launch_bounds__(THREADS, 1)
void topo_extrema_kernel(const float* __restrict__ xin,
                         const float* __restrict__ yin,
                         float* __restrict__ lists)
{
    const int tid   = threadIdx.x;
    const int imgId = blockIdx.x;                      // 0..255 = x, 256..511 = y
    const float* src = ((imgId < 256) ? xin : yin) + (size_t)(imgId & 255) * NPIX;

    extern __shared__ __align__(16) unsigned char smem_raw[];
    float*    simg   = (float*)smem_raw;               // 65536 f32  (256 KB)
    unsigned* sminM  = (unsigned*)(simg + NPIX);       // 2048 words (8 KB)
    unsigned* smaxM  = sminM + 2048;                   // 2048 words (8 KB)
    unsigned* shist8 = smaxM + 2048;                   // 2048 words (8 KB, 8 banks)
    unsigned* shist  = shist8 + 2048;                  // 256 words
    unsigned* spfx   = shist + 256;                    // 256 words
    unsigned* scand  = spfx + 256;                     // 128 words
    unsigned* sctrl  = scand + 128;                    // 8 words

    // gfx1250 prefetch of this block's image (global_prefetch_b8)
    __builtin_prefetch(src + (size_t)tid * PPT, 0, 3);

    // ---- stage image into LDS: read each global byte exactly once ----
#if ASYNC_LDS
    {
        const char* g = (const char*)src;
        char*       l = (char*)simg;
#pragma unroll
        for (int it = 0; it < 16; ++it) {
            const size_t off = (size_t)it * 16384u + (size_t)tid * 16u;
            __builtin_amdgcn_global_load_async_to_lds_b128(
                to_global_v4(g + off), to_lds_v4(l + off), 0, 0);
        }
    }
    WAIT_ASYNC();
    __syncthreads();
#else
    {
        const float4* g4 = (const float4*)src;
        float4*       l4 = (float4*)simg;
#pragma unroll
        for (int it = 0; it < 16; ++it) {
            const int idx = it * THREADS + tid;        // coalesced b128
            l4[idx] = g4[idx];
        }
    }
    __syncthreads();
#endif

    // ---- 4-neighbor extrema masks (boundary = +/-BIG, exactly like reference) ----
    {
        const int base = tid * PPT;
        unsigned w0min = 0, w1min = 0, w0max = 0, w1max = 0;
        for (int k = 0; k < PPT; ++k) {
            const int p = base + k;
            const int r = p >> 8, c = p & 255;
            const float f  = simg[p];
            const float up = (r > 0)   ? simg[p - 256] : BIGF;
            const float dn = (r < 255) ? simg[p + 256] : BIGF;
            const float lf = (c > 0)   ? simg[p - 1]   : BIGF;
            const float rt = (c < 255) ? simg[p + 1]   : BIGF;
            const float nmin = fminf(fminf(up, dn), fminf(lf, rt));
            const float upx = (r > 0)   ? up : -BIGF;
            const float dnx = (r < 255) ? dn : -BIGF;
            const float lfx = (c > 0)   ? lf : -BIGF;
            const float rtx = (c < 255) ? rt : -BIGF;
            const float nmax = fmaxf(fmaxf(upx, dnx), fmaxf(lfx, rtx));
            if (f <= nmin) { if (k < 32) w0min |= 1u << k; else w1min |= 1u << (k - 32); }
            if (f >= nmax) { if (k < 32) w0max |= 1u << k; else w1max |= 1u << (k - 32); }
        }
        sminM[2 * tid]     = w0min;
        sminM[2 * tid + 1] = w1min;
        smaxM[2 * tid]     = w0max;
        smaxM[2 * tid + 1] = w1max;
    }
    __syncthreads();

    float* out = lists + (size_t)imgId * (4 * K_PD);
    // b0: K smallest minima (asc, pad +BIG)     d0: K smallest maxima (asc, pad +BIG)
    // b1: K largest  maxima (desc, pad -BIG)    d1: K largest  minima (desc, pad -BIG)
    topk_select(simg, sminM, 0,  BIGF, out + 0 * K_PD, shist8, shist, spfx, scand, sctrl, tid);
    topk_select(simg, smaxM, 0,  BIGF, out + 1 * K_PD, shist8, shist, spfx, scand, sctrl, tid);
    topk_select(simg, smaxM, 1, -BIGF, out + 2 * K_PD, shist8, shist, spfx, scand, sctrl, tid);
    topk_select(simg, sminM, 1, -BIGF, out + 3 * K_PD, shist8, shist, spfx, scand, sctrl, tid);
}

// ---------- kernel B: per-sample diagram distance ------------------------------
__global__ __launch_bounds__(K_PD)
void topo_loss_kernel(const float* __restrict__ lists, float* __restrict__ losses)
{
    __shared__ float red[K_PD];
    const int b = blockIdx.x;      // sample 0..255
    const int i = threadIdx.x;     // slot 0..127
    const float* lx = lists + (size_t)b * (4 * K_PD);
    const float* ly = lists + (size_t)(b + 256) * (4 * K_PD);

    const float b0x = lx[i], d0x = lx[K_PD + i], b1x = lx[2 * K_PD + i], d1x = lx[3 * K_PD + i];
    const float b0y = ly[i], d0y = ly[K_PD + i], b1y = ly[2 * K_PD + i], d1y = ly[3 * K_PD + i];

    const bool v0x = (b0x < HBIGF) && (d0x < HBIGF);
    const float px00 = v0x ? b0x : 0.0f;
    const float px01 = v0x ? fmaxf(d0x, b0x) : 0.0f;
    const bool v1x = (b1x > -HBIGF) && (d1x > -HBIGF);
    const float px10 = v1x ? b1x : 0.0f;
    const float px11 = v1x ? fminf(d1x, b1x) : 0.0f;

    const bool v0y = (b0y < HBIGF) && (d0y < HBIGF);
    const float py00 = v0y ? b0y : 0.0f;
    const float py01 = v0y ? fmaxf(d0y, b0y) : 0.0f;
    const bool v1y = (b1y > -HBIGF) && (d1y > -HBIGF);
    const float py10 = v1y ? b1y : 0.0f;
    const float py11 = v1y ? fminf(d1y, b1y) : 0.0f;

    const float e0 = px00 - py00, e1 = px01 - py01, e2 = px10 - py10, e3 = px11 - py11;
    red[i] = e0 * e0 + e1 * e1 + e2 * e2 + e3 * e3;
    __syncthreads();
    for (int s = K_PD / 2; s > 0; s >>= 1) {
        if (i < s) red[i] += red[i + s];
        __syncthreads();
    }
    if (i == 0) losses[b] = red[0];
}

// ---------- kernel C: deterministic mean over the batch ------------------------
__global__ __launch_bounds__(256)
void topo_mean_kernel(const float* __restrict__ losses, float* __restrict__ out)
{
    __shared__ float red[256];
    const int i = threadIdx.x;
    red[i] = losses[i];
    __syncthreads();
    for (int s = 128; s > 0; s >>= 1) {
        if (i < s) red[i] += red[i + s];
        __syncthreads();
    }
    if (i == 0) out[0] = red[0] * (1.0f / 256.0f);
}

// ---------- launch -------------------------------------------------------------
extern "C" void kernel_launch(void* const* d_in, const int* in_sizes, int n_in,
                              void* d_out, int out_size, void* d_ws, size_t ws_size,
                              hipStream_t stream)
{
    (void)in_sizes; (void)n_in; (void)out_size; (void)ws_size;
    const float* x = (const float*)d_in[0];   // inputs  [256,256,256] f32
    const float* y = (const float*)d_in[1];   // targets [256,256,256] f32
    float* lists  = (float*)d_ws;             // 512 images * 4 lists * 128 f32 = 1 MB
    float* losses = lists + 512 * 4 * K_PD;   // 256 f32
    float* out    = (float*)d_out;            // scalar f32

    // img(256KB) + 2 masks(16KB) + hist8(8KB) + hist(1KB) + pfx(1KB) + cand(512B) + ctrl(32B)
    const int smem_bytes = NPIX * 4 + 2 * 2048 * 4 + 2048 * 4 + 256 * 4 + 256 * 4
                         + 128 * 4 + 8 * 4;
    (void)hipFuncSetAttribute(reinterpret_cast<const void*>(&topo_extrema_kernel),
                              hipFuncAttributeMaxDynamicSharedMemorySize, smem_bytes);

    topo_extrema_kernel<<<512, THREADS, smem_bytes, stream>>>(x, y, lists);
    topo_loss_kernel<<<256, K_PD, 0, stream>>>(lists, losses);
    topo_mean_kernel<<<1, 256, 0, stream>>>(losses, out);
}